// LearnableSparseHandler_47682726921014
// MI455X (gfx1250) — compile-verified
//
#include <hip/hip_runtime.h>
#include <math.h>

// Problem dims (fixed by reference setup_inputs)
#define B_  4
#define T_  12
#define C_  64
#define H_  128
#define W_  128
#define N_  (H_ * W_)          // 16384
#define KTOP 8192              // N * (1 - 0.5)
#define EPSC 1e-5f
#define SLOPE 0.01f

typedef float v2f __attribute__((ext_vector_type(2)));
typedef float v8f __attribute__((ext_vector_type(8)));

__device__ __forceinline__ float leaky(float v) { return v > 0.0f ? v : SLOPE * v; }

// ---------------------------------------------------------------------------
// K1: temporal mean + max reduce.  x (B,T,C,H,W) -> feat (B, 128, N)
//     channels [0,64) = mean over T, channels [64,128) = max over T
// ---------------------------------------------------------------------------
__global__ __launch_bounds__(256) void reduce_mean_max(const float* __restrict__ x,
                                                       float* __restrict__ feat) {
  const int n4 = N_ / 4;                       // float4 granularity
  long tid = (long)blockIdx.x * blockDim.x + threadIdx.x;  // over B*C*n4
  int bc = (int)(tid / n4);
  int nv = (int)(tid % n4);
  int b = bc / C_, c = bc % C_;
  const float4* xp = (const float4*)x + ((long)b * T_ * C_ + c) * n4 + nv;
  const long strideT = (long)C_ * n4;
  float4 v = xp[0];
  float4 s = v, mx = v;
  for (int t = 1; t < T_; ++t) {
    float4 u = xp[t * strideT];
    s.x += u.x; s.y += u.y; s.z += u.z; s.w += u.w;
    mx.x = fmaxf(mx.x, u.x); mx.y = fmaxf(mx.y, u.y);
    mx.z = fmaxf(mx.z, u.z); mx.w = fmaxf(mx.w, u.w);
  }
  const float inv = 1.0f / (float)T_;
  float4 mean = {s.x * inv, s.y * inv, s.z * inv, s.w * inv};
  float4* fm = (float4*)feat + ((long)b * 128 + c) * n4 + nv;
  float4* fx = (float4*)feat + ((long)b * 128 + 64 + c) * n4 + nv;
  *fm = mean;
  *fx = mx;
}

// ---------------------------------------------------------------------------
// K2: 1x1 conv 128 -> 32 as GEMM via V_WMMA_F32_16X16X4_F32.
//     D(16 pix x 32 outC) += A(16 pix x 4 ch) * B(4 ch x 16 outC) (x2 N-tiles)
//     Fragment layout per ISA 7.12.2 (32-bit A 16x4 / C-D 16x16 analogs):
//       per lane, element j of a fragment register pair = K index k0+2*half+j
//       (half = lane>=16); C/D VGPR i holds rows M=i (lanes 0-15) / M=i+8.
// ---------------------------------------------------------------------------
__global__ __launch_bounds__(256) void conv1_wmma(const float* __restrict__ feat,
                                                  const float* __restrict__ w1,
                                                  const float* __restrict__ b1,
                                                  float* __restrict__ h1pre) {
  __shared__ float sw[32 * 128];               // w1[o][c], 16 KB
  for (int i = threadIdx.x; i < 32 * 128; i += 256) sw[i] = w1[i];
  __syncthreads();

  const int wave = threadIdx.x >> 5;
  const int lane = threadIdx.x & 31;
  const int hl = lane >> 4;                    // half-wave
  const int m  = lane & 15;

  const int p0 = (blockIdx.x * 8 + wave) * 16; // global pixel base (16-aligned, within one batch)
  const int b  = p0 / N_;
  const int n0 = p0 % N_;
  const float* fb = feat + (long)b * 128 * N_;

  v8f acc0 = {0,0,0,0,0,0,0,0};
  v8f acc1 = {0,0,0,0,0,0,0,0};

  for (int k0 = 0; k0 < 128; k0 += 4) {
    const int ka = k0 + 2 * hl;
    v2f a;
    a.x = fb[(long)ka * N_ + n0 + m];
    a.y = fb[(long)(ka + 1) * N_ + n0 + m];
    v2f bA, bB;
    bA.x = sw[m * 128 + ka];        bA.y = sw[m * 128 + ka + 1];
    bB.x = sw[(m + 16) * 128 + ka]; bB.y = sw[(m + 16) * 128 + ka + 1];
    acc0 = __builtin_amdgcn_wmma_f32_16x16x4_f32(false, a, false, bA, (short)0, acc0, false, false);
    acc1 = __builtin_amdgcn_wmma_f32_16x16x4_f32(false, a, false, bB, (short)0, acc1, false, false);
  }

  const float bias0 = b1[m], bias1 = b1[m + 16];
  float* out = h1pre + (long)b * 32 * N_;
  for (int i = 0; i < 8; ++i) {
    const int pix = n0 + i + 8 * hl;
    out[(long)m * N_ + pix]        = acc0[i] + bias0;
    out[(long)(m + 16) * N_ + pix] = acc1[i] + bias1;
  }
}

// ---------------------------------------------------------------------------
// K3: GroupNorm statistics. One block per (batch, group); 8 ch x 16384 px each.
// ---------------------------------------------------------------------------
__global__ __launch_bounds__(256) void gn_stats(const float* __restrict__ h1pre,
                                                float* __restrict__ stats) {
  const int bg = blockIdx.x;                   // 0..15
  const int b = bg / 4, g = bg % 4;
  const float* base = h1pre + ((long)b * 32 + g * 8) * N_;
  const long total = 8L * N_;                  // 131072
  float s = 0.0f, ss = 0.0f;
  for (long i = threadIdx.x; i < total; i += 256) {
    float v = base[i];
    s += v; ss += v * v;
  }
  __shared__ float rs[256], rss[256];
  rs[threadIdx.x] = s; rss[threadIdx.x] = ss;
  __syncthreads();
  for (int o = 128; o > 0; o >>= 1) {
    if (threadIdx.x < o) { rs[threadIdx.x] += rs[threadIdx.x + o]; rss[threadIdx.x] += rss[threadIdx.x + o]; }
    __syncthreads();
  }
  if (threadIdx.x == 0) {
    float mu = rs[0] / (float)total;
    float var = rss[0] / (float)total - mu * mu;
    stats[bg * 2 + 0] = mu;
    stats[bg * 2 + 1] = rsqrtf(var + EPSC);
  }
}

// ---------------------------------------------------------------------------
// K4: GroupNorm apply + LeakyReLU.  h1pre -> h1 (B,32,N)
// ---------------------------------------------------------------------------
__global__ __launch_bounds__(256) void gn_apply(const float* __restrict__ h1pre,
                                                const float* __restrict__ stats,
                                                const float* __restrict__ gw,
                                                const float* __restrict__ gb,
                                                float* __restrict__ h1) {
  long i = (long)blockIdx.x * 256 + threadIdx.x;   // over B*32*N
  int cN = (int)(i / N_);
  int b = cN / 32, c = cN % 32;
  int g = c / 8;
  float mu = stats[(b * 4 + g) * 2 + 0];
  float rstd = stats[(b * 4 + g) * 2 + 1];
  float v = (h1pre[i] - mu) * rstd * gw[c] + gb[c];
  h1[i] = leaky(v);
}

// ---------------------------------------------------------------------------
// K5: 3x3 conv 32 -> 16 (pad 1) as implicit GEMM via WMMA f32 16x16x4.
//     K-dim = 9 taps x 32 ch, stepped by 4.  16-pixel tiles never wrap a row
//     (W=128 is a multiple of 16); per-lane zero fill handles padding.
// ---------------------------------------------------------------------------
__global__ __launch_bounds__(256) void conv2_wmma(const float* __restrict__ h1,
                                                  const float* __restrict__ w2,
                                                  const float* __restrict__ b2,
                                                  float* __restrict__ h2) {
  __shared__ float sw[16 * 32 * 9];            // w2[o][c][kh][kw], 18 KB
  for (int i = threadIdx.x; i < 16 * 32 * 9; i += 256) sw[i] = w2[i];
  __syncthreads();

  const int wave = threadIdx.x >> 5;
  const int lane = threadIdx.x & 31;
  const int hl = lane >> 4;
  const int m  = lane & 15;

  const int p0 = (blockIdx.x * 8 + wave) * 16;
  const int b  = p0 / N_;
  const int n0 = p0 % N_;
  const int h  = n0 / W_;
  const int w0 = n0 % W_;
  const float* ib = h1 + (long)b * 32 * N_;

  v8f acc = {0,0,0,0,0,0,0,0};

  for (int kh = -1; kh <= 1; ++kh) {
    const int hh = h + kh;
    const bool rowok = (hh >= 0) && (hh < H_);
    for (int kw = -1; kw <= 1; ++kw) {
      const int cc = w0 + m + kw;              // this lane's input column
      const bool ok = rowok && (cc >= 0) && (cc < W_);
      const long aoff = (long)hh * W_ + cc;
      const int widx = (kh + 1) * 3 + (kw + 1);
      for (int c0 = 0; c0 < 32; c0 += 4) {
        const int ka = c0 + 2 * hl;
        v2f a, bb;
        a.x = ok ? ib[(long)ka * N_ + aoff] : 0.0f;
        a.y = ok ? ib[(long)(ka + 1) * N_ + aoff] : 0.0f;
        bb.x = sw[(m * 32 + ka) * 9 + widx];
        bb.y = sw[(m * 32 + ka + 1) * 9 + widx];
        acc = __builtin_amdgcn_wmma_f32_16x16x4_f32(false, a, false, bb, (short)0, acc, false, false);
      }
    }
  }

  const float bias = b2[m];
  float* out = h2 + (long)b * 16 * N_;
  for (int i = 0; i < 8; ++i) {
    out[(long)m * N_ + n0 + i + 8 * hl] = leaky(acc[i] + bias);
  }
}

// ---------------------------------------------------------------------------
// K6: 1x1 conv 16 -> 1 + sigmoid -> scores (B, N)
// ---------------------------------------------------------------------------
__global__ __launch_bounds__(256) void score_kernel(const float* __restrict__ h2,
                                                    const float* __restrict__ w3,
                                                    const float* __restrict__ b3,
                                                    float* __restrict__ scores) {
  int i = blockIdx.x * 256 + threadIdx.x;      // over B*N = 65536
  int b = i / N_, n = i % N_;
  const float* base = h2 + (long)b * 16 * N_ + n;
  float z = b3[0];
  for (int c = 0; c < 16; ++c) z += base[(long)c * N_] * w3[c];
  scores[i] = 1.0f / (1.0f + __expf(-z));
}

// ---------------------------------------------------------------------------
// K7: exact top-K membership per batch.  Sigmoid outputs are positive floats,
//     so uint bit-pattern order == float order.  31-step greedy bit search
//     finds the K-th largest value; ties at the threshold admit lowest
//     indices first (matches jax.lax.top_k).
// ---------------------------------------------------------------------------
__global__ __launch_bounds__(1024) void topk_mask(const float* __restrict__ scores,
                                                  int* __restrict__ mask) {
  __shared__ int scnt;
  const int b = blockIdx.x;
  const unsigned* sd = (const unsigned*)(scores + (long)b * N_);

  unsigned thr = 0;
  for (int bit = 30; bit >= 0; --bit) {
    unsigned cand = thr | (1u << bit);
    if (threadIdx.x == 0) scnt = 0;
    __syncthreads();
    int local = 0;
    for (int i = threadIdx.x; i < N_; i += 1024) local += (sd[i] >= cand) ? 1 : 0;
    atomicAdd(&scnt, local);
    __syncthreads();
    if (scnt >= KTOP) thr = cand;              // all threads see same scnt
    __syncthreads();
  }
  // count strictly greater than threshold value
  if (threadIdx.x == 0) scnt = 0;
  __syncthreads();
  int local = 0;
  for (int i = threadIdx.x; i < N_; i += 1024) local += (sd[i] > thr) ? 1 : 0;
  atomicAdd(&scnt, local);
  __syncthreads();
  const int remaining = KTOP - scnt;           // ties to accept (lowest idx first)

  int* mb = mask + (long)b * N_;
  for (int i = threadIdx.x; i < N_; i += 1024) {
    const unsigned v = sd[i];
    int keep = 0;
    if (v > thr) keep = 1;
    else if (v == thr) {
      int rank = 0;                            // ties are rare with random floats
      for (int j = 0; j < i; ++j) rank += (sd[j] == thr) ? 1 : 0;
      keep = (rank < remaining) ? 1 : 0;
    }
    mb[i] = keep;
  }
}

// ---------------------------------------------------------------------------
// K8: out = mask ? x : 0, float4-vectorized (forward STE gate == 1.0)
// ---------------------------------------------------------------------------
__global__ __launch_bounds__(256) void apply_mask(const float* __restrict__ x,
                                                  const int* __restrict__ mask,
                                                  float* __restrict__ out) {
  const int n4 = N_ / 4;
  long i = (long)blockIdx.x * 256 + threadIdx.x;   // over B*T*C*n4
  long btc = i / n4;
  int nv = (int)(i % n4);
  int b = (int)(btc / (T_ * C_));
  const int4 mk = ((const int4*)(mask + (long)b * N_))[nv];
  const float4 v = ((const float4*)x)[i];
  float4 o;
  o.x = mk.x ? v.x : 0.0f;
  o.y = mk.y ? v.y : 0.0f;
  o.z = mk.z ? v.z : 0.0f;
  o.w = mk.w ? v.w : 0.0f;
  ((float4*)out)[i] = o;
}

// ---------------------------------------------------------------------------
extern "C" void kernel_launch(void* const* d_in, const int* in_sizes, int n_in,
                              void* d_out, int out_size, void* d_ws, size_t ws_size,
                              hipStream_t stream) {
  (void)in_sizes; (void)n_in; (void)out_size; (void)ws_size;
  const float* x    = (const float*)d_in[0];
  const float* w1   = (const float*)d_in[1];
  const float* b1   = (const float*)d_in[2];
  const float* gn_w = (const float*)d_in[3];
  const float* gn_b = (const float*)d_in[4];
  const float* w2   = (const float*)d_in[5];
  const float* b2   = (const float*)d_in[6];
  const float* w3   = (const float*)d_in[7];
  const float* b3   = (const float*)d_in[8];
  float* out = (float*)d_out;

  char* ws = (char*)d_ws;
  // workspace layout (bytes); h1/h2 reuse feat's region once feat is dead
  float* feat   = (float*)(ws + 0);                       // 4*128*16384*4 = 33554432
  float* h1pre  = (float*)(ws + 33554432);                // 4*32*16384*4  =  8388608
  float* h1     = (float*)(ws + 0);                       // reuse feat region
  float* h2     = (float*)(ws + 16777216);                // reuse feat region (disjoint from h1)
  float* scores = (float*)(ws + 41943040);                // 4*16384*4     =   262144
  int*   mask   = (int*)  (ws + 42205184);                // 4*16384*4     =   262144
  float* stats  = (float*)(ws + 42467328);                // 16*2 floats

  const int n4 = N_ / 4;

  // K1: temporal reduce (mean+max)  -> feat
  reduce_mean_max<<<(B_ * C_ * n4) / 256, 256, 0, stream>>>(x, feat);

  // K2: 1x1 conv via WMMA           -> h1pre
  conv1_wmma<<<(B_ * N_) / 128, 256, 0, stream>>>(feat, w1, b1, h1pre);

  // K3/K4: GroupNorm + LeakyReLU    -> h1
  gn_stats<<<B_ * 4, 256, 0, stream>>>(h1pre, stats);
  gn_apply<<<(B_ * 32 * N_) / 256, 256, 0, stream>>>(h1pre, stats, gn_w, gn_b, h1);

  // K5: 3x3 conv via WMMA + leaky   -> h2
  conv2_wmma<<<(B_ * N_) / 128, 256, 0, stream>>>(h1, w2, b2, h2);

  // K6: 1x1 conv + sigmoid          -> scores
  score_kernel<<<(B_ * N_) / 256, 256, 0, stream>>>(h2, w3, b3, scores);

  // K7: exact top-K membership      -> mask
  topk_mask<<<B_, 1024, 0, stream>>>(scores, mask);

  // K8: masked copy                 -> out
  apply_mask<<<(long)(B_ * T_ * C_ * n4) / 256, 256, 0, stream>>>(x, mask, out);
}